// CrossAttention_5677946765857
// MI455X (gfx1250) — compile-verified
//
#include <hip/hip_runtime.h>
#include <math.h>

// ---------------- problem constants ----------------
#define BATCH   2
#define SEQ     3072
#define DMODEL  320
#define HEADS   8
#define DH      40
#define DHP     48                   // head dim padded to 3x16 tiles
#define BH      (BATCH*HEADS)        // 16
#define NJT     (SEQ/16)             // 192 key chunks / row tiles per seq
#define SCALE   0.15811388300841897f // 40^-0.5
#define NEGMAX  -3.402823466e38f
#define CHUNKF  (16*DHP)             // 768 floats per K or V chunk

typedef float v2f __attribute__((ext_vector_type(2)));
typedef float v8f __attribute__((ext_vector_type(8)));
typedef unsigned int v4u __attribute__((ext_vector_type(4)));
typedef int v4i __attribute__((ext_vector_type(4)));
typedef int v8i __attribute__((ext_vector_type(8)));

#if __has_include(<hip/amd_detail/amd_gfx1250_TDM.h>)
#define TDM_SIX_ARGS 1
#else
#define TDM_SIX_ARGS 0
#endif

// D = A(16x4) * B(4x16) + C, fp32 WMMA (CDNA5 V_WMMA_F32_16X16X4_F32)
__device__ __forceinline__ v8f wmma4(v2f a, v2f b, v8f c) {
  return __builtin_amdgcn_wmma_f32_16x16x4_f32(
      false, a, false, b, (short)0, c, false, false);
}

// Tensor Data Mover: contiguous 1-row tile of `nelem` f32 from global -> LDS.
// D# packed per CDNA5 ISA §8.3/8.4: count=1, data_size=4B, tile_dim0=nelem,
// tile_dim1=1, tensor_dim0=nelem, type=2 ("image").
__device__ __forceinline__ void tdm_load_f32(unsigned lds_off, const float* gsrc,
                                             unsigned nelem) {
  unsigned long long ga = (unsigned long long)(uintptr_t)gsrc;
  v4u g0;
  g0.x = 1u;                                           // count=1 (valid, user)
  g0.y = lds_off;                                      // lds_addr (bytes)
  g0.z = (unsigned)(ga & 0xFFFFFFFFu);                 // global_addr[31:0]
  g0.w = (unsigned)((ga >> 32) & 0x1FFFFFFu) | (2u << 30); // addr[56:32]|type=2
  v8i g1;
  g1[0] = (int)(2u << 16);                             // data_size = 4B
  g1[1] = (int)((nelem & 0xFFFFu) << 16);              // tensor_dim0[15:0]
  g1[2] = (int)((nelem >> 16) | (1u << 16));           // dim0[31:16], dim1=1
  g1[3] = (int)((nelem & 0xFFFFu) << 16);              // tile_dim0 = nelem
  g1[4] = 1;                                           // tile_dim1=1, tile_dim2=0
  g1[5] = (int)nelem;                                  // tensor_dim0_stride lo
  g1[6] = 0;
  g1[7] = 0;
  v4i z = {0, 0, 0, 0};
#if TDM_SIX_ARGS
  v8i z8 = {0, 0, 0, 0, 0, 0, 0, 0};
  __builtin_amdgcn_tensor_load_to_lds(g0, g1, z, z, z8, 0);
#else
  __builtin_amdgcn_tensor_load_to_lds(g0, g1, z, z, 0);
#endif
}

// ---------------- mask kernel ----------------
// nearest resize 512x384 -> 64x48 is exact stride-8 sampling
__global__ void mask_kernel(const float* __restrict__ mask1,
                            const float* __restrict__ mask2,
                            int* __restrict__ m1b, int* __restrict__ m2b) {
  int idx = blockIdx.x * blockDim.x + threadIdx.x;
  if (idx >= BATCH * SEQ) return;
  int b = idx / SEQ, i = idx % SEQ;
  size_t off = (size_t)b * 512 * 384 + (size_t)(i / 48) * 8 * 384 + (size_t)(i % 48) * 8;
  m1b[idx] = (mask1[off] >= 0.5f) ? 1 : 0;
  m2b[idx] = (mask2[off] >= 0.5f) ? 1 : 0;
}

// ---------------- QKV projection ----------------
__global__ void proj_kernel(const float* __restrict__ x,
                            const float* __restrict__ ctx,
                            const float* __restrict__ Wq,
                            const float* __restrict__ Wk,
                            const float* __restrict__ Wv,
                            float* __restrict__ Qws,
                            float* __restrict__ Kws,
                            float* __restrict__ Vws) {
  int which = blockIdx.y;
  const float* A = (which == 0) ? x : ctx;
  const float* W = (which == 0) ? Wq : ((which == 1) ? Wk : Wv);
  float* Dst     = (which == 0) ? Qws : ((which == 1) ? Kws : Vws);

  int wave = (blockIdx.x * blockDim.x + threadIdx.x) >> 5;
  int lane = threadIdx.x & 31;
  int mt = wave / (DMODEL / 16);
  int ct = wave % (DMODEL / 16);
  int half = lane >> 4, l16 = lane & 15;

  int arow = mt * 16 + l16;
  int col  = ct * 16 + l16;

  v8f acc = {};
  for (int kk = 0; kk < DMODEL; kk += 4) {
    int k0 = kk + half * 2;
    v2f a = *(const v2f*)(A + (size_t)arow * DMODEL + k0);
    v2f bf;
    bf.x = W[(size_t)k0 * DMODEL + col];
    bf.y = W[(size_t)(k0 + 1) * DMODEL + col];
    acc = wmma4(a, bf, acc);
  }
#pragma unroll
  for (int r = 0; r < 8; ++r) {
    int m = mt * 16 + r + 8 * half;
    int b = m / SEQ, n = m % SEQ;
    int h = col / DH, dcol = col % DH;
    Dst[((size_t)(b * HEADS + h) * SEQ + n) * DHP + dcol] = acc[r];
  }
}

// ---------------- flash attention ----------------
// Block = 8 waves sharing one head; wave w handles query rows [rb*16, rb*16+16).
// K/V 16-key chunks staged into LDS by the Tensor Data Mover (double-buffered),
// so all 8 waves read shared ds data instead of 8x redundant global traffic.
__global__ void attn_kernel(const float* __restrict__ Qws,
                            const float* __restrict__ Kws,
                            const float* __restrict__ Vws,
                            const int* __restrict__ m1b,
                            const int* __restrict__ m2b,
                            float* __restrict__ Ows) {
  __shared__ float ldsK[2][CHUNKF];        // 2 x 3 KB
  __shared__ float ldsV[2][CHUNKF];        // 2 x 3 KB
  __shared__ float ldsP[8 * 256];          // per-wave 16x16 P staging

  int wib  = threadIdx.x >> 5;
  int lane = threadIdx.x & 31;
  int h  = blockIdx.x / (NJT / 8);         // 0..15 (b*HEADS + head)
  int rb = (blockIdx.x % (NJT / 8)) * 8 + wib;
  int b  = h / HEADS;
  int half = lane >> 4, l16 = lane & 15;
  float* P = ldsP + wib * 256;

  const float* Qh = Qws + (size_t)h * SEQ * DHP;
  const float* Kh = Kws + (size_t)h * SEQ * DHP;
  const float* Vh = Vws + (size_t)h * SEQ * DHP;

  // Q fragments (A-layout, K-dim 40 -> 10 steps)
  v2f qf[10];
  {
    const float* qp = Qh + (size_t)(rb * 16 + l16) * DHP + half * 2;
#pragma unroll
    for (int s = 0; s < 10; ++s) qf[s] = *(const v2f*)(qp + s * 4);
  }

  unsigned rowmask = 0;
  {
    int rbase = rb * 16 + half * 8;
#pragma unroll
    for (int r = 0; r < 8; ++r)
      rowmask |= (m1b[b * SEQ + rbase + r] ? 1u : 0u) << r;
  }

  float mrow[8], lrow[8];
#pragma unroll
  for (int r = 0; r < 8; ++r) { mrow[r] = NEGMAX; lrow[r] = 0.f; }
  v8f o0 = {}, o1 = {}, o2 = {};

  // prologue: stage chunk 0 into buffer 0
  if (wib == 0) {
    tdm_load_f32((unsigned)(uintptr_t)&ldsK[0][0], Kh, CHUNKF);
    tdm_load_f32((unsigned)(uintptr_t)&ldsV[0][0], Vh, CHUNKF);
  }

  for (int jb = 0; jb < NJT; ++jb) {
    int cur = jb & 1;
    if (wib == 0) {
      if (jb + 1 < NJT) {               // prefetch next chunk (other buffer)
        int nxt = (jb + 1) & 1;
        tdm_load_f32((unsigned)(uintptr_t)&ldsK[nxt][0],
                     Kh + (size_t)(jb + 1) * CHUNKF, CHUNKF);
        tdm_load_f32((unsigned)(uintptr_t)&ldsV[nxt][0],
                     Vh + (size_t)(jb + 1) * CHUNKF, CHUNKF);
        __builtin_amdgcn_s_wait_tensorcnt(2);  // chunk jb (2 oldest ops) done
      } else {
        __builtin_amdgcn_s_wait_tensorcnt(0);
      }
    }
    __syncthreads();                    // publish buf[cur] to all waves

    const float* Kt = &ldsK[cur][0];
    const float* Vt = &ldsV[cur][0];

    int jcol = jb * 16 + l16;
    int m2j = m2b[b * SEQ + jcol];

    // S = Q * K^T : B frag B[d][j] = K[j][d], contiguous pair (ds_load_b64)
    v8f s = {};
    {
      const float* kp = Kt + (size_t)l16 * DHP + half * 2;
#pragma unroll
      for (int st = 0; st < 10; ++st) {
        v2f kf = *(const v2f*)(kp + st * 4);
        s = wmma4(qf[st], kf, s);
      }
    }

    // scale, mask, online softmax (row = r + 8*half, col = l16)
#pragma unroll
    for (int r = 0; r < 8; ++r) {
      float t = s[r] * SCALE;
      if (m2j && ((rowmask >> r) & 1u)) t = NEGMAX;
      float mx = t;
      mx = fmaxf(mx, __shfl_xor(mx, 1, 32));
      mx = fmaxf(mx, __shfl_xor(mx, 2, 32));
      mx = fmaxf(mx, __shfl_xor(mx, 4, 32));
      mx = fmaxf(mx, __shfl_xor(mx, 8, 32));
      float nm    = fmaxf(mrow[r], mx);
      float alpha = __expf(mrow[r] - nm);
      float p     = __expf(t - nm);
      float su = p;
      su += __shfl_xor(su, 1, 32);
      su += __shfl_xor(su, 2, 32);
      su += __shfl_xor(su, 4, 32);
      su += __shfl_xor(su, 8, 32);
      lrow[r] = lrow[r] * alpha + su;
      mrow[r] = nm;
      o0[r] *= alpha; o1[r] *= alpha; o2[r] *= alpha;
      P[(r + 8 * half) * 16 + l16] = p;
    }
    asm volatile("s_wait_dscnt 0" ::: "memory");

    // O += P(16x16) * V(16x48), P reloaded in A-layout from LDS
#pragma unroll
    for (int st = 0; st < 4; ++st) {
      int k0 = st * 4 + half * 2;
      v2f a = *(const v2f*)(P + l16 * 16 + k0);
      const float* v0p = Vt + (size_t)k0 * DHP;
      const float* v1p = v0p + DHP;
      v2f vb0, vb1, vb2;
      vb0.x = v0p[l16];      vb0.y = v1p[l16];
      vb1.x = v0p[16 + l16]; vb1.y = v1p[16 + l16];
      vb2.x = v0p[32 + l16]; vb2.y = v1p[32 + l16];
      o0 = wmma4(a, vb0, o0);
      o1 = wmma4(a, vb1, o1);
      o2 = wmma4(a, vb2, o2);
    }
    __syncthreads();                    // all reads of buf[cur] done -> reusable
  }

  // epilogue: normalize and merge heads -> Ows[b][n][h_local*40 + d]
  int hl = h % HEADS;
#pragma unroll
  for (int r = 0; r < 8; ++r) {
    float inv = 1.0f / lrow[r];
    int n = rb * 16 + r + 8 * half;
    float* orow = Ows + ((size_t)b * SEQ + n) * DMODEL + hl * DH;
    orow[l16]      = o0[r] * inv;
    orow[16 + l16] = o1[r] * inv;
    if (l16 < 8) orow[32 + l16] = o2[r] * inv;
  }
}

// ---------------- output projection ----------------
__global__ void outproj_kernel(const float* __restrict__ Ows,
                               const float* __restrict__ Wo,
                               const float* __restrict__ bo,
                               float* __restrict__ out) {
  int wave = (blockIdx.x * blockDim.x + threadIdx.x) >> 5;
  int lane = threadIdx.x & 31;
  int mt = wave / (DMODEL / 16);
  int ct = wave % (DMODEL / 16);
  int half = lane >> 4, l16 = lane & 15;
  int arow = mt * 16 + l16;
  int col  = ct * 16 + l16;

  v8f acc = {};
  for (int kk = 0; kk < DMODEL; kk += 4) {
    int k0 = kk + half * 2;
    v2f a = *(const v2f*)(Ows + (size_t)arow * DMODEL + k0);
    v2f bf;
    bf.x = Wo[(size_t)k0 * DMODEL + col];
    bf.y = Wo[(size_t)(k0 + 1) * DMODEL + col];
    acc = wmma4(a, bf, acc);
  }
  float bias = bo[col];
#pragma unroll
  for (int r = 0; r < 8; ++r) {
    int m = mt * 16 + r + 8 * half;
    out[(size_t)m * DMODEL + col] = acc[r] + bias;
  }
}

// ---------------- launcher ----------------
extern "C" void kernel_launch(void* const* d_in, const int* in_sizes, int n_in,
                              void* d_out, int out_size, void* d_ws, size_t ws_size,
                              hipStream_t stream) {
  const float* x     = (const float*)d_in[0];
  const float* ctx   = (const float*)d_in[1];
  const float* mask1 = (const float*)d_in[2];
  const float* mask2 = (const float*)d_in[3];
  const float* Wq    = (const float*)d_in[4];
  const float* Wk    = (const float*)d_in[5];
  const float* Wv    = (const float*)d_in[6];
  const float* Wo    = (const float*)d_in[7];
  const float* bo    = (const float*)d_in[8];
  float* out = (float*)d_out;

  const size_t perQKV = (size_t)BH * SEQ * DHP;
  float* Qws = (float*)d_ws;
  float* Kws = Qws + perQKV;
  float* Vws = Kws + perQKV;
  float* Ows = Vws + perQKV;
  int*   m1b = (int*)(Ows + (size_t)BATCH * SEQ * DMODEL);
  int*   m2b = m1b + BATCH * SEQ;

  mask_kernel<<<dim3((BATCH * SEQ + 255) / 256), 256, 0, stream>>>(mask1, mask2, m1b, m2b);
  proj_kernel<<<dim3(960, 3), 256, 0, stream>>>(x, ctx, Wq, Wk, Wv, Qws, Kws, Vws);
  // 16 heads * 24 row-block-groups = 384 blocks, 8 waves each
  attn_kernel<<<dim3(384), 256, 0, stream>>>(Qws, Kws, Vws, m1b, m2b, Ows);
  outproj_kernel<<<dim3(960), 256, 0, stream>>>(Ows, Wo, bo, out);
}